// Classifier_22651657519678
// MI455X (gfx1250) — compile-verified
//
#include <hip/hip_runtime.h>
#include <hip/hip_bf16.h>

typedef float v2f __attribute__((ext_vector_type(2)));
typedef float v8f __attribute__((ext_vector_type(8)));

#define TPB 256

// ---------------------------------------------------------------------------
// Generic dense layer: Y[M,Co] = bn(act(X[M,Ci] @ W^T + b))   (W is (Co,Ci))
// One wave per 16x16 output tile, fp32 WMMA 16x16x4 along Ci.
// REQUIRES: M % 16 == 0 (true for every call site). Co may be arbitrary.
// ---------------------------------------------------------------------------
__global__ __launch_bounds__(TPB) void dense_wmma_kernel(
    const float* __restrict__ X, int ldX,
    const float* __restrict__ W,
    const float* __restrict__ b,
    const float* __restrict__ g,
    const float* __restrict__ be,
    float* __restrict__ Y, int ldY,
    int M, int Ci, int Co, int act, int tilesM, int tilesN)
{
    int wid = blockIdx.x * (TPB / 32) + (threadIdx.x >> 5);
    if (wid >= tilesM * tilesN) return;          // wave-uniform: EXEC stays full
    int tileM = wid % tilesM;
    int tileN = wid / tilesM;

    int lane = threadIdx.x & 31;
    int half = lane >> 4;       // 0: K pair {0,1}, 1: K pair {2,3}
    int l    = lane & 15;
    int row  = tileM * 16 + l;                 // always < M (M % 16 == 0)
    int col  = tileN * 16 + l;
    int colc = (col < Co) ? col : (Co - 1);    // clamp: garbage cols never stored

    const float* Xr = X + (size_t)row  * ldX;
    const float* Wr = W + (size_t)colc * Ci;
    int kbase = half * 2;

    v8f acc = {};
    int kq = Ci & ~3;                          // full quads: unconditional loads
    for (int kk = 0; kk < kq; kk += 4) {
        v2f a, bf;
        a.x  = Xr[kk + kbase];
        a.y  = Xr[kk + kbase + 1];
        bf.x = Wr[kk + kbase];
        bf.y = Wr[kk + kbase + 1];
        acc = __builtin_amdgcn_wmma_f32_16x16x4_f32(
                  false, a, false, bf, (short)0, acc, false, false);
    }
    if (kq < Ci) {                             // tail (only Ci=3 here)
        int ka0 = kq + kbase;
        int ka1 = ka0 + 1;
        int c0 = (ka0 < Ci) ? ka0 : (Ci - 1);  // clamped unconditional loads
        int c1 = (ka1 < Ci) ? ka1 : (Ci - 1);
        float xa0 = Xr[c0], xa1 = Xr[c1];
        float wb0 = Wr[c0], wb1 = Wr[c1];
        v2f a, bf;
        a.x  = (ka0 < Ci) ? xa0 : 0.0f;        // post-load select -> v_cndmask
        a.y  = (ka1 < Ci) ? xa1 : 0.0f;
        bf.x = (ka0 < Ci) ? wb0 : 0.0f;
        bf.y = (ka1 < Ci) ? wb1 : 0.0f;
        acc = __builtin_amdgcn_wmma_f32_16x16x4_f32(
                  false, a, false, bf, (short)0, acc, false, false);
    }

    // C/D layout: VGPR v -> M = tileM*16 + v + half*8, N = tileN*16 + l
    #pragma unroll
    for (int v = 0; v < 8; ++v) {
        int m = tileM * 16 + v + half * 8;
        int n = tileN * 16 + l;
        if (n < Co) {
            float y = acc[v] + b[n];
            if (act) y = fmaxf(y, 0.0f);
            if (g)   y = y * g[n] + be[n];
            Y[(size_t)m * ldY + n] = y;
        }
    }
}

// ---------------------------------------------------------------------------
__global__ void fill_zero_kernel(float* p, int n) {
    int i = blockIdx.x * blockDim.x + threadIdx.x;
    if (i < n) p[i] = 0.0f;
}

// rep[n,p,:] = x[n, sidx[p], :]
__global__ void gather_rep_kernel(const float* __restrict__ x,
                                  const int* __restrict__ sidx,
                                  float* __restrict__ rep,
                                  int N, int Pout, int M)
{
    int i = blockIdx.x * blockDim.x + threadIdx.x;
    if (i >= N * Pout) return;
    int n = i / Pout, p = i % Pout;
    int j = sidx[p];
    const float* s = x + ((size_t)n * M + j) * 3;
    rep[i * 3 + 0] = s[0];
    rep[i * 3 + 1] = s[1];
    rep[i * 3 + 2] = s[2];
}

// Dilated kNN dropping self: select K*D+1 nearest, keep indices 1, 1+D, ...
#define MAXSEL 73
__global__ void knn_kernel(const float* __restrict__ rep,
                           const float* __restrict__ pts,
                           int* __restrict__ idx,
                           int N, int P, int M, int K, int D)
{
    int i = blockIdx.x * blockDim.x + threadIdx.x;
    if (i >= N * P) return;
    int n = i / P;
    const float* r = rep + (size_t)i * 3;
    float rx = r[0], ry = r[1], rz = r[2];
    const float* pt = pts + (size_t)n * M * 3;

    int nsel = K * D + 1;
    float dist[MAXSEL];
    int   ind [MAXSEL];
    for (int t = 0; t < nsel; ++t) { dist[t] = 3.4e38f; ind[t] = 0; }

    for (int m = 0; m < M; ++m) {
        float dx = pt[m * 3 + 0] - rx;
        float dy = pt[m * 3 + 1] - ry;
        float dz = pt[m * 3 + 2] - rz;
        float d2 = dx * dx + dy * dy + dz * dz;
        if (d2 < dist[nsel - 1]) {
            int pos = nsel - 1;
            while (pos > 0 && dist[pos - 1] > d2) {
                dist[pos] = dist[pos - 1];
                ind [pos] = ind [pos - 1];
                --pos;
            }
            dist[pos] = d2;
            ind [pos] = m;
        }
    }
    for (int k = 0; k < K; ++k) idx[(size_t)i * K + k] = ind[1 + k * D];
}

// loc[n,p,k,:] = pts[n, idx[n,p,k], :] - rep[n,p,:]
__global__ void gather_loc_kernel(const float* __restrict__ rep,
                                  const float* __restrict__ pts,
                                  const int* __restrict__ idx,
                                  float* __restrict__ loc,
                                  int N, int P, int M, int K)
{
    int i = blockIdx.x * blockDim.x + threadIdx.x;
    if (i >= N * P * K) return;
    int p_np = i / K;
    int n = p_np / P;
    int j = idx[i];
    const float* s = pts + ((size_t)n * M + j) * 3;
    const float* r = rep + (size_t)p_np * 3;
    loc[(size_t)i * 3 + 0] = s[0] - r[0];
    loc[(size_t)i * 3 + 1] = s[1] - r[1];
    loc[(size_t)i * 3 + 2] = s[2] - r[2];
}

// cat[n,p,k, cmid + c] = fts_l[n, idx[n,p,k], c],  c < cf  (cf, cmid mult. of 4)
__global__ void gather_fts_kernel(const float* __restrict__ ftsl,
                                  const int* __restrict__ idx,
                                  float* __restrict__ cat,
                                  int N, int P, int M, int K,
                                  int cf, int cmid, int csep)
{
    int i = blockIdx.x * blockDim.x + threadIdx.x;
    if (i >= N * P * K) return;
    int n = (i / K) / P;
    int j = idx[i];
    const float4* s = (const float4*)(ftsl + ((size_t)n * M + j) * cf);
    float4* d = (float4*)(cat + (size_t)i * csep + cmid);
    int nq = cf >> 2;
    for (int c = 0; c < nq; ++c) d[c] = s[c];
}

// t[np,o] = relu( sum_{k,d} loc[np,k,d] * xcW[o,d,k] + xcb[o] )
__global__ void xconv_kernel(const float* __restrict__ loc,
                             const float* __restrict__ xcW,
                             const float* __restrict__ xcb,
                             float* __restrict__ t,
                             int NP, int K, int K2)
{
    int i = blockIdx.x * blockDim.x + threadIdx.x;
    if (i >= NP * K2) return;
    int o = i % K2, np = i / K2;
    const float* L = loc + (size_t)np * K * 3;
    const float* Wp = xcW + (size_t)o * 3 * K;
    float s = xcb[o];
    for (int k = 0; k < K; ++k)
        for (int d = 0; d < 3; ++d)
            s += L[k * 3 + d] * Wp[d * K + k];
    t[(size_t)np * K2 + o] = fmaxf(s, 0.0f);
}

// fX[np,k,c] = sum_j Xb[np, k*K + j] * cat[np, j, c]
__global__ void ftsX_kernel(const float* __restrict__ Xb,
                            const float* __restrict__ cat,
                            float* __restrict__ fX,
                            int NP, int K, int K2, int csep)
{
    int i = blockIdx.x * blockDim.x + threadIdx.x;
    if (i >= NP * K * csep) return;
    int c  = i % csep;
    int k  = (i / csep) % K;
    int np = i / (csep * K);
    const float* Xr = Xb + (size_t)np * K2 + k * K;
    const float* C  = cat + (size_t)np * K * csep;
    float s = 0.0f;
    for (int j = 0; j < K; ++j) s += Xr[j] * C[(size_t)j * csep + c];
    fX[i] = s;
}

// mid[np, c*dm+m] = sum_k fX[np,k,c] * dwW[(c*dm+m)*K + k] + dwb[c*dm+m]
__global__ void dw_kernel(const float* __restrict__ fX,
                          const float* __restrict__ dwW,
                          const float* __restrict__ dwb,
                          float* __restrict__ mid,
                          int NP, int K, int csep, int dm)
{
    int i = blockIdx.x * blockDim.x + threadIdx.x;
    if (i >= NP * csep * dm) return;
    int m  = i % dm;
    int c  = (i / dm) % csep;
    int np = i / (dm * csep);
    int cm = c * dm + m;
    float s = dwb[cm];
    const float* f = fX + (size_t)np * K * csep + c;
    const float* w = dwW + (size_t)cm * K;
    for (int k = 0; k < K; ++k) s += f[(size_t)k * csep] * w[k];
    mid[(size_t)np * csep * dm + cm] = s;
}

// out[n,c] = mean_p logits[n,p,c]
__global__ void mean_kernel(const float* __restrict__ logits,
                            float* __restrict__ out,
                            int N, int P, int C)
{
    int i = blockIdx.x * blockDim.x + threadIdx.x;
    if (i >= N * C) return;
    int n = i / C, c = i % C;
    float s = 0.0f;
    for (int p = 0; p < P; ++p) s += logits[((size_t)n * P + p) * C + c];
    out[i] = s / (float)P;
}

// ---------------------------------------------------------------------------
extern "C" void kernel_launch(void* const* d_in, const int* in_sizes, int n_in,
                              void* d_out, int out_size, void* d_ws, size_t ws_size,
                              hipStream_t stream)
{
    (void)in_sizes; (void)n_in; (void)out_size; (void)ws_size;

    const float* x    = (const float*)d_in[0];
    const int*   sidx = (const int*)d_in[1];
    // params flattened in JAX pytree order: 'fcn' (sorted keys W,b,be,g) then
    // 'layers' (per-layer keys sorted: d1,d2,dw_W,dw_b,in,pw,xc_W,xc_b,xd1,xd2)
    auto LW = [&](int li, int off) { return (const float*)d_in[12 + 24 * li + off]; };
    auto FP = [&](int off)         { return (const float*)d_in[2 + off]; };

    const int N = 16;
    float* ws = (float*)d_ws;
    size_t off = 0;
    auto alloc = [&](size_t nf) { float* p = ws + off; off += nf; return p; };
    float* ftsA  = alloc((size_t)16384 * 160);
    float* ftsB  = alloc((size_t)16384 * 160);
    float* ftsl  = alloc((size_t)16384 * 80);
    float* locb  = alloc((size_t)131072 * 3);
    float* catb  = alloc((size_t)131072 * 72);
    float* fXb   = alloc((size_t)131072 * 72);
    float* t1    = alloc((size_t)1048576);
    float* t2    = alloc((size_t)1048576);
    float* midb  = alloc((size_t)16384 * 144);
    float* repb  = alloc((size_t)16 * 120 * 3);
    float* h1    = alloc((size_t)1920 * 128);
    float* h2    = alloc((size_t)1920 * 64);
    float* logit = alloc((size_t)1920 * 40);
    int*   idxb  = (int*)alloc((size_t)131072);

    auto dense = [&](const float* Xp, int ldX, const float* Wp, const float* bp,
                     const float* gp, const float* bep, float* Yp, int ldY,
                     int M, int Ci, int Co, bool act) {
        int tilesM = (M + 15) / 16, tilesN = (Co + 15) / 16;
        int waves  = tilesM * tilesN;
        int blocks = (waves + (TPB / 32) - 1) / (TPB / 32);
        dense_wmma_kernel<<<blocks, TPB, 0, stream>>>(
            Xp, ldX, Wp, bp, gp, bep, Yp, ldY, M, Ci, Co, act ? 1 : 0, tilesM, tilesN);
    };
    auto gridOf = [](long n) { return (int)((n + TPB - 1) / TPB); };

    // layer configs: cin, cout, K, D, dm
    const int cins[5]  = {3, 32, 64, 96, 128};
    const int couts[5] = {32, 64, 96, 128, 160};
    const int Ks[5]    = {8, 8, 8, 12, 12};
    const int Ds[5]    = {1, 2, 4, 4, 6};
    const int dms[5]   = {4, 2, 2, 2, 2};

    // fts starts as zeros_like(x): (N,1024,3)
    fill_zero_kernel<<<gridOf(16384 * 3), TPB, 0, stream>>>(ftsA, 16384 * 3);

    int P = 1024;
    const float* pts = x;
    float* ftsCur = ftsA;
    float* ftsNext = ftsB;

    for (int i = 0; i < 5; ++i) {
        int cin = cins[i], cout = couts[i], K = Ks[i], D = Ds[i], dm = dms[i];
        int cf = cout / 2, cmid = cout / 4, csep = cf + cmid, K2 = K * K;
        int Pout = (i == 3) ? 120 : P;
        int NPo = N * Pout;

        const float* rep = pts;
        if (i == 3) {
            gather_rep_kernel<<<gridOf(N * 120), TPB, 0, stream>>>(x, sidx, repb, N, 120, 1024);
            rep = repb;
        }

        // fts_l = dense(in)(fts)  over P (pre-downsample) rows
        dense(ftsCur, cin, LW(i, 10), LW(i, 11), LW(i, 13), LW(i, 12),
              ftsl, cf, N * P, cin, cf, true);

        knn_kernel<<<(NPo + 63) / 64, 64, 0, stream>>>(rep, pts, idxb, N, Pout, P, K, D);
        gather_loc_kernel<<<gridOf((long)NPo * K), TPB, 0, stream>>>(rep, pts, idxb, locb, N, Pout, P, K);
        gather_fts_kernel<<<gridOf((long)NPo * K), TPB, 0, stream>>>(ftsl, idxb, catb, N, Pout, P, K, cf, cmid, csep);

        // lifted = d2(d1(loc)) -> cat[..., 0:cmid]
        dense(locb, 3,    LW(i, 0), LW(i, 1), LW(i, 3), LW(i, 2), fXb,  cmid, NPo * K, 3,    cmid, true);
        dense(fXb,  cmid, LW(i, 4), LW(i, 5), LW(i, 7), LW(i, 6), catb, csep, NPo * K, cmid, cmid, true);

        // X-transform
        xconv_kernel<<<gridOf((long)NPo * K2), TPB, 0, stream>>>(locb, LW(i, 18), LW(i, 19), t1, NPo, K, K2);
        dense(t1, K2, LW(i, 20), LW(i, 21), nullptr, nullptr, t2, K2, NPo, K2, K2, true);
        dense(t2, K2, LW(i, 22), LW(i, 23), nullptr, nullptr, t1, K2, NPo, K2, K2, false);

        // fts_X = X @ fts_cat ; depthwise ; pointwise
        ftsX_kernel<<<gridOf((long)NPo * K * csep), TPB, 0, stream>>>(t1, catb, fXb, NPo, K, K2, csep);
        dw_kernel<<<gridOf((long)NPo * csep * dm), TPB, 0, stream>>>(fXb, LW(i, 8), LW(i, 9), midb, NPo, K, csep, dm);
        dense(midb, csep * dm, LW(i, 14), LW(i, 15), LW(i, 17), LW(i, 16),
              ftsNext, cout, NPo, csep * dm, cout, true);

        pts = rep;
        P = Pout;
        float* tmp = ftsCur; ftsCur = ftsNext; ftsNext = tmp;
    }

    // FCN head over (N*120, 160)
    dense(ftsCur, 160, FP(0), FP(1), FP(3), FP(2), h1, 128, 1920, 160, 128, true);
    dense(h1,     128, FP(4), FP(5), FP(7), FP(6), h2, 64,  1920, 128, 64,  true);
    dense(h2,     64,  FP(8), FP(9), nullptr, nullptr, logit, 40, 1920, 64, 40, false);

    mean_kernel<<<gridOf(16 * 40), TPB, 0, stream>>>(logit, (float*)d_out, 16, 120, 40);
}